// LatentTokenDecoder_5351529251372
// MI455X (gfx1250) — compile-verified
//
#include <hip/hip_runtime.h>

typedef float v2f __attribute__((ext_vector_type(2)));
typedef float v8f __attribute__((ext_vector_type(8)));

#define BDIM 256
#define NWAVES (BDIM / 32)

// One workgroup per sample. Fused: style FC -> modulation -> demod ->
// 3x3 conv as fp32 WMMA GEMM (im2col over zero-padded, channel-interleaved
// LDS input; K ordered tap-major so inner-loop addressing folds to aligned
// b64 loads with immediate offsets) -> bilinear 2x upsample fused store.
template <int CIN, int COUT, int H>
__global__ __launch_bounds__(BDIM) void style_conv_layer(
    const float* __restrict__ x,       // [B,32]
    const float* __restrict__ fc_w,    // [32,32]
    const float* __restrict__ fc_b,    // [32]
    const float* __restrict__ in,      // [B,CIN,H,H] or broadcast const
    long long in_bstride,              // 0 for broadcast const input
    const float* __restrict__ conv_w,  // [COUT,CIN,3,3]
    const float* __restrict__ mod_w,   // [CIN,32]
    const float* __restrict__ mod_b,   // [CIN]
    float* __restrict__ out)           // [B,COUT,2H,2H]
{
    constexpr int W   = H;
    constexpr int HW  = H * W;          // power of two (16/64/256/1024)
    constexpr int PW  = W + 2;
    constexpr int PHW = (H + 2) * PW;   // padded plane (pixels)
    constexpr int K   = CIN * 9;        // 288 / 144 / 72 / 36
    constexpr int AS  = K + 2;          // even row stride -> aligned b64 loads
    constexpr int NT  = HW / 16;        // 16-pixel N tiles

    __shared__ float s_style[32];
    __shared__ float s_mod[CIN];
    __shared__ float s_d[16];
    __shared__ __align__(16) float s_A[16 * AS];      // weights, k = q*CIN + c
    __shared__ __align__(16) float s_inp[PHW * CIN];  // padded, ch-interleaved
    __shared__ float s_y[COUT * HW];                  // conv out pre-upsample

    const int tid = threadIdx.x;
    const int b   = blockIdx.x;

    // ---- style = x[b] @ fc_w^T + fc_b ----
    if (tid < 32) {
        float acc = fc_b[tid];
        const float* xr = x + (long long)b * 32;
        const float* wr = fc_w + tid * 32;
        #pragma unroll
        for (int j = 0; j < 32; ++j) acc += xr[j] * wr[j];
        s_style[tid] = acc;
    }
    // ---- zero the padded interleaved input ----
    for (int i = tid; i < PHW * CIN; i += BDIM) s_inp[i] = 0.0f;
    __syncthreads();

    // ---- s = style @ mod_w^T + mod_b ----
    if (tid < CIN) {
        float acc = mod_b[tid];
        const float* wr = mod_w + tid * 32;
        #pragma unroll
        for (int j = 0; j < 32; ++j) acc += s_style[j] * wr[j];
        s_mod[tid] = acc;
    }
    // ---- stage input: coalesced global read -> interleaved padded LDS ----
    for (int i = tid; i < CIN * HW; i += BDIM) {
        const int c = i / HW;           // HW, W powers of two -> shifts
        const int r = i % HW;
        const int y = r / W, xx = r % W;
        s_inp[((y + 1) * PW + (xx + 1)) * CIN + c] = in[in_bstride * b + i];
    }
    __syncthreads();

    // ---- modulate weights into zero-padded 16 x K A (k = q*CIN + c) ----
    for (int i = tid; i < 16 * K; i += BDIM) {
        const int o = i / K, r = i % K;
        const int q = r / CIN, c = r % CIN;   // CIN power of two -> shifts
        float v = 0.0f;
        if (o < COUT) v = conv_w[(o * CIN + c) * 9 + q] * s_mod[c];
        s_A[o * AS + r] = v;
    }
    __syncthreads();

    // ---- demodulate: row *= rsqrt(sum(row^2) + 1e-8) ----
    if (tid < 16) {
        float sum = 0.0f;
        for (int r = 0; r < K; ++r) { const float v = s_A[tid * AS + r]; sum += v * v; }
        s_d[tid] = rsqrtf(sum + 1e-8f);
    }
    __syncthreads();
    for (int i = tid; i < COUT * K; i += BDIM) {
        const int o = i / K, r = i % K;
        s_A[o * AS + r] *= s_d[o];
    }
    __syncthreads();

    // ---- GEMM: D[16x16] = A[16x4] x B[4x16] over K (fp32 WMMA) ----
    // A frag (16x4 f32): lane m=lane&15; v0/v1 = K 0,1 (lanes 0-15) / 2,3 (16-31)
    // B frag mirrored; C/D: VGPR r -> M = r + (lane<16?0:8), N = lane&15.
    const int wave  = tid >> 5;
    const int lane  = tid & 31;
    const int m     = lane & 15;
    const int khalf = (lane < 16) ? 0 : 2;
    const int abase = m * AS + khalf;          // even; +kk immediate per step

    for (int t = wave; t < NT; t += NWAVES) {  // wave-uniform: EXEC stays all-1
        const int p  = t * 16 + m;             // output pixel for this B column
        const int py = p / W, px = p % W;
        // per-lane interleaved B base (even); +compile-time offset per kk
        const int bbase = (py * PW + px) * CIN + khalf;

        v8f acc = {0.f, 0.f, 0.f, 0.f, 0.f, 0.f, 0.f, 0.f};
        #pragma unroll
        for (int kk = 0; kk < K; kk += 4) {
            const int q    = kk / CIN;         // constants after unroll
            const int cb   = kk % CIN;
            const int boff = ((q / 3) * PW + (q % 3)) * CIN + cb;
            const v2f a  = *(const v2f*)(s_A + abase + kk);     // ds_load_b64
            const v2f bb = *(const v2f*)(s_inp + bbase + boff); // ds_load_b64
            acc = __builtin_amdgcn_wmma_f32_16x16x4_f32(
                false, a, false, bb, (short)0, acc, false, false);
        }
        #pragma unroll
        for (int r = 0; r < 8; ++r) {
            const int o = r + ((lane < 16) ? 0 : 8);
            if (o < COUT) s_y[o * HW + t * 16 + m] = acc[r];
        }
    }
    __syncthreads();

    // ---- fused bilinear 2x upsample (half-pixel centers, clamped) + store ----
    float* ob = out + (long long)b * COUT * 4 * HW;
    for (int i = tid; i < COUT * 4 * HW; i += BDIM) {
        const int o   = i / (4 * HW);
        const int rem = i % (4 * HW);
        const int oy  = rem / (2 * W);
        const int ox  = rem % (2 * W);
        const float sy = oy * 0.5f - 0.25f;
        const float sx = ox * 0.5f - 0.25f;
        const float fy = floorf(sy), fx = floorf(sx);
        const float wy = sy - fy,    wx = sx - fx;
        const int y0 = (int)fy, x0 = (int)fx;
        const int y0c = y0 < 0 ? 0 : y0;
        const int y1c = (y0 + 1 > H - 1) ? H - 1 : y0 + 1;
        const int x0c = x0 < 0 ? 0 : x0;
        const int x1c = (x0 + 1 > W - 1) ? W - 1 : x0 + 1;
        const float* yb = s_y + o * HW;
        const float v00 = yb[y0c * W + x0c], v01 = yb[y0c * W + x1c];
        const float v10 = yb[y1c * W + x0c], v11 = yb[y1c * W + x1c];
        ob[i] = (1.f - wy) * ((1.f - wx) * v00 + wx * v01) +
                wy        * ((1.f - wx) * v10 + wx * v11);
    }
}

extern "C" void kernel_launch(void* const* d_in, const int* in_sizes, int n_in,
                              void* d_out, int out_size, void* d_ws, size_t ws_size,
                              hipStream_t stream)
{
    (void)in_sizes; (void)n_in; (void)out_size; (void)d_ws; (void)ws_size;

    const float* x    = (const float*)d_in[0];
    const float* fc_w = (const float*)d_in[1];
    const float* fc_b = (const float*)d_in[2];
    const float* cst  = (const float*)d_in[3];
    const float* cw0  = (const float*)d_in[4];
    const float* mw0  = (const float*)d_in[5];
    const float* mb0  = (const float*)d_in[6];
    const float* cw1  = (const float*)d_in[7];
    const float* mw1  = (const float*)d_in[8];
    const float* mb1  = (const float*)d_in[9];
    const float* cw2  = (const float*)d_in[10];
    const float* mw2  = (const float*)d_in[11];
    const float* mb2  = (const float*)d_in[12];
    const float* cw3  = (const float*)d_in[13];
    const float* mw3  = (const float*)d_in[14];
    const float* mb3  = (const float*)d_in[15];

    constexpr int B = 4096;
    float* f0 = (float*)d_out;                        // [B,16, 8, 8]
    float* f1 = f0 + (size_t)B * 16 * 8 * 8;          // [B, 8,16,16]
    float* f2 = f1 + (size_t)B * 8 * 16 * 16;         // [B, 4,32,32]
    float* f3 = f2 + (size_t)B * 4 * 32 * 32;         // [B, 4,64,64]

    dim3 grid(B), block(BDIM);
    style_conv_layer<32, 16,  4><<<grid, block, 0, stream>>>(
        x, fc_w, fc_b, cst, 0,                      cw0, mw0, mb0, f0);
    style_conv_layer<16,  8,  8><<<grid, block, 0, stream>>>(
        x, fc_w, fc_b, f0, (long long)16 * 8 * 8,   cw1, mw1, mb1, f1);
    style_conv_layer< 8,  4, 16><<<grid, block, 0, stream>>>(
        x, fc_w, fc_b, f1, (long long)8 * 16 * 16,  cw2, mw2, mb2, f2);
    style_conv_layer< 4,  4, 32><<<grid, block, 0, stream>>>(
        x, fc_w, fc_b, f2, (long long)4 * 32 * 32,  cw3, mw3, mb3, f3);
}